// CapsuleLayer_30279519437231
// MI455X (gfx1250) — compile-verified
//
#include <hip/hip_runtime.h>
#include <hip/hip_bf16.h>

// Problem constants (from reference setup_inputs)
#define B_   256
#define I_   1152
#define Q_   8
#define J_   10
#define P_   16

typedef __attribute__((ext_vector_type(2))) float     v2f;
typedef __attribute__((ext_vector_type(8))) float     v8f;
typedef __attribute__((ext_vector_type(4))) _Float16  v4h;
typedef __attribute__((ext_vector_type(4))) unsigned int u32x4;

// ---------------------------------------------------------------------------
// Kernel A: ihat[b,i,j,p] = sum_q W[j,i,p,q] * x[b,i,q]  via fp32 WMMA.
// Per (j,i): GEMM tile M=16 batches, N=16 p, K=8 q -> two V_WMMA_F32_16X16X4_F32.
// Grid: (16 b-tiles, 10 j, 9 i-blocks), 256 thr = 8 waves, each wave 16 i's.
// Output stored fp16 (halves routing-pass HBM traffic), layout [b][i][j][p]
// so the routing kernel streams 320B-contiguous rows per i.
// ---------------------------------------------------------------------------
__global__ __launch_bounds__(256) void capsule_ihat_wmma(
    const float* __restrict__ x,      // [B, I, Q]
    const float* __restrict__ W,      // [J, I, P, Q]
    _Float16*    __restrict__ ihat)   // [B, I, J, P] fp16
{
  __shared__ __align__(16) _Float16 lds[8][264];   // 16x16 halves per wave (+pad)

  const int btile = blockIdx.x;        // 0..15
  const int j     = blockIdx.y;        // 0..9
  const int iblk  = blockIdx.z;        // 0..8
  const int wave  = threadIdx.x >> 5;
  const int lane  = threadIdx.x & 31;
  const int m     = lane & 15;         // A: M row (batch); B: N col (p)
  const int hi    = lane >> 4;         // K half select (lanes 16-31 hold K=2,3)
  const int b0    = btile << 4;
  const int i0    = iblk * 128 + wave * 16;

  const float* xrow  = x + (size_t)(b0 + m) * I_ * Q_ + 2 * hi;
  const size_t wbase = (size_t)j * I_ * P_ * Q_ + (size_t)m * Q_ + 2 * hi;

  for (int ii = 0; ii < 16; ++ii) {
    const int i = i0 + ii;
    const float* xp = xrow + (size_t)i * Q_;
    const float* wp = W + wbase + (size_t)i * (P_ * Q_);

    // A[m, k]: k = 2*hi + {0,1}  (first wmma q=0..3, second q=4..7)
    v2f a0 = *(const v2f*)xp;
    v2f a1 = *(const v2f*)(xp + 4);
    // B[k, p=m] = W[j,i,p,q=k]
    v2f w0 = *(const v2f*)wp;
    v2f w1 = *(const v2f*)(wp + 4);

    v8f acc = {0.f, 0.f, 0.f, 0.f, 0.f, 0.f, 0.f, 0.f};
    acc = __builtin_amdgcn_wmma_f32_16x16x4_f32(false, a0, false, w0,
                                                (short)0, acc, false, false);
    acc = __builtin_amdgcn_wmma_f32_16x16x4_f32(false, a1, false, w1,
                                                (short)0, acc, false, false);

    // D layout: lane holds N=m, rows M = r + 8*hi. Transpose via LDS so we can
    // emit 16B contiguous fp16 stores per batch row.
    #pragma unroll
    for (int r = 0; r < 8; ++r)
      lds[wave][(hi * 8 + r) * 16 + m] = (_Float16)acc[r];
    asm volatile("s_wait_dscnt 0x0" ::: "memory");   // CDNA5 split DS counter

    const int row = lane >> 1;   // batch row within tile
    const int hq  = lane & 1;    // which 8-p half
    u32x4 v = *(const u32x4*)&lds[wave][row * 16 + hq * 8];
    _Float16* op = ihat + (((size_t)(b0 + row) * I_ + i) * J_ + j) * P_ + hq * 8;
    *(u32x4*)op = v;
  }
}

// ---------------------------------------------------------------------------
// Kernel C: one routing iteration, fully fused per batch.
//   logits bb[j,i] = o_cum[j,:] . ihat[b,i,j,:]   (o_cum = sum of prior outputs,
//                                                  == reference's b state)
//   c = softmax_j(bb);  s[j,p] = sum_i c[j,i]*ihat[b,i,j,p];  o = squash(s)
// Grid: 256 blocks (one per b), 512 thr = 128 quads; quad owns one i per step
// (lane owns 4 p's -> coalesced 32B per j). Deterministic shuffle+LDS tree
// reduction (no float atomics -> bit-stable across graph replays).
// ---------------------------------------------------------------------------
__global__ __launch_bounds__(512) void capsule_route(
    const _Float16* __restrict__ ihat,   // [B, I, J, P] fp16
    float*          __restrict__ o_cum,  // [B, J, P]
    float*          __restrict__ out,    // [B, J, P]
    int update_ocum, int write_out)
{
  __shared__ float oc[J_ * P_];
  __shared__ float wp_lds[J_ * P_ * 16];

  const int b = blockIdx.x;
  const int t = threadIdx.x;
  if (t < J_ * P_) oc[t] = o_cum[b * (J_ * P_) + t];
  __syncthreads();

  const int quad = t >> 2;   // 0..127 -> i index stride
  const int qq   = t & 3;    // lane within quad -> p group [4*qq, 4*qq+3]

  float sp[J_][4];
  #pragma unroll
  for (int j = 0; j < J_; ++j)
    #pragma unroll
    for (int p = 0; p < 4; ++p) sp[j][p] = 0.f;

  const _Float16* base = ihat + (size_t)b * I_ * (J_ * P_);
  for (int k = 0; k < 9; ++k) {                 // 9 * 128 quads = 1152 i
    const int i = k * 128 + quad;
    const _Float16* row = base + (size_t)i * (J_ * P_);
    if (k < 8)
      __builtin_prefetch(row + 128 * (J_ * P_), 0, 0);   // global_prefetch_b8

    float hv[J_][4];
    float bb[J_];
    #pragma unroll
    for (int j = 0; j < J_; ++j) {
      v4h h = *(const v4h*)(row + j * P_ + qq * 4);
      float d = 0.f;
      #pragma unroll
      for (int p = 0; p < 4; ++p) {
        hv[j][p] = (float)h[p];
        d += oc[j * P_ + qq * 4 + p] * hv[j][p];
      }
      bb[j] = d;
    }
    #pragma unroll
    for (int j = 0; j < J_; ++j) {              // reduce dot over the quad
      bb[j] += __shfl_xor(bb[j], 1, 32);
      bb[j] += __shfl_xor(bb[j], 2, 32);
    }
    // softmax over J (each lane redundantly, all-lane uniform within quad)
    float mx = bb[0];
    #pragma unroll
    for (int j = 1; j < J_; ++j) mx = fmaxf(mx, bb[j]);
    float e[J_];
    float esum = 0.f;
    #pragma unroll
    for (int j = 0; j < J_; ++j) { e[j] = __expf(bb[j] - mx); esum += e[j]; }
    const float inv = 1.f / esum;
    #pragma unroll
    for (int j = 0; j < J_; ++j) {
      const float c = e[j] * inv;
      #pragma unroll
      for (int p = 0; p < 4; ++p) sp[j][p] += c * hv[j][p];
    }
  }

  // Reduce s over the 8 quads of each wave (fixed-order shuffle tree).
  #pragma unroll
  for (int j = 0; j < J_; ++j)
    #pragma unroll
    for (int p = 0; p < 4; ++p) {
      sp[j][p] += __shfl_xor(sp[j][p], 4, 32);
      sp[j][p] += __shfl_xor(sp[j][p], 8, 32);
      sp[j][p] += __shfl_xor(sp[j][p], 16, 32);
    }
  const int w = t >> 5;
  if ((t & 31) < 4) {                 // lanes 0..3 hold wave totals per p-group
    #pragma unroll
    for (int j = 0; j < J_; ++j)
      #pragma unroll
      for (int p = 0; p < 4; ++p)
        wp_lds[(j * P_ + qq * 4 + p) * 16 + w] = sp[j][p];
  }
  __syncthreads();

  // Final deterministic sum over 16 waves + fused squash + state update.
  if (t < J_ * P_) {
    float s = 0.f;
    #pragma unroll
    for (int ww = 0; ww < 16; ++ww) s += wp_lds[t * 16 + ww];
    float s2 = s * s;                 // norm over the 16 p's of this j
    s2 += __shfl_xor(s2, 1, 32);
    s2 += __shfl_xor(s2, 2, 32);
    s2 += __shfl_xor(s2, 4, 32);
    s2 += __shfl_xor(s2, 8, 32);
    const float scale = s2 / (1.f + s2) * rsqrtf(s2 + 1e-7f);
    const float o = scale * s;
    if (update_ocum) o_cum[b * (J_ * P_) + t] = oc[t] + o;
    if (write_out)   out[b * (J_ * P_) + t] = o;
  }
}

__global__ void capsule_zero(float* __restrict__ p, int n) {
  int t = blockIdx.x * blockDim.x + threadIdx.x;
  if (t < n) p[t] = 0.f;
}

// ---------------------------------------------------------------------------
// Launch: zero o_cum; WMMA ihat; 3 fused routing sweeps (iter 0 gets uniform
// softmax automatically from o_cum==0). Workspace: fp16 ihat (94.4MB) + o_cum.
// ---------------------------------------------------------------------------
extern "C" void kernel_launch(void* const* d_in, const int* in_sizes, int n_in,
                              void* d_out, int out_size, void* d_ws, size_t ws_size,
                              hipStream_t stream) {
  const float* x = (const float*)d_in[0];   // [256,1152,8]
  const float* W = (const float*)d_in[1];   // [10,1152,16,8]
  float* out = (float*)d_out;               // [256,10,16]

  const size_t ihat_bytes = (size_t)B_ * I_ * J_ * P_ * sizeof(_Float16);
  _Float16* ihat = (_Float16*)d_ws;
  float* o_cum = (float*)((char*)d_ws + ihat_bytes);

  const int oc_n = B_ * J_ * P_;
  capsule_zero<<<(oc_n + 255) / 256, 256, 0, stream>>>(o_cum, oc_n);

  capsule_ihat_wmma<<<dim3(B_ / 16, J_, I_ / 128), 256, 0, stream>>>(x, W, ihat);

  // r = 0, 1: update o_cum; r = 2: write final output
  capsule_route<<<B_, 512, 0, stream>>>(ihat, o_cum, out, 1, 0);
  capsule_route<<<B_, 512, 0, stream>>>(ihat, o_cum, out, 1, 0);
  capsule_route<<<B_, 512, 0, stream>>>(ihat, o_cum, out, 0, 1);
}